// BERTIdealEmissionRateCompressionModule_20753281974886
// MI455X (gfx1250) — compile-verified
//
#include <hip/hip_runtime.h>

// Problem constants from the reference: L,B,H,S,D = 12,8,12,512,768
#define L_ 12
#define B_ 8
#define H_ 12
#define S_ 512

typedef __attribute__((ext_vector_type(2))) float v2f;
typedef __attribute__((ext_vector_type(8))) float v8f;

// Kernel 1: y_soft[b,s] = -prod_l ( mean_h att[l,b,h,0,s] )
// One wave (32 lanes) computes 16 consecutive s values for one batch b.
// Head-sum via V_WMMA_F32_16X16X4_F32 with an all-ones A matrix:
//   D[m,n] = C[m,n] + sum_k B[k,n]  for every row m,
// which is invariant to the exact K-row placement inside the B operand.
// B operand fill: vgpr0 = heads {g*4+0 (lanes 0-15), g*4+1 (lanes 16-31)},
//                 vgpr1 = heads {g*4+2, g*4+3}; column N = lane & 15.
// Three accumulating K=4 WMMAs sum all 12 heads in exact f32.
__global__ __launch_bounds__(32) void ysoft_wmma_kernel(const float* __restrict__ att,
                                                        float* __restrict__ ysoft) {
    const int lane = threadIdx.x;        // 0..31, full wave, no divergence before WMMA
    const int n    = lane & 15;          // column within 16-wide s tile
    const int half = lane >> 4;          // 0 or 1: selects head within a pair
    const int b    = blockIdx.y;
    const int s    = blockIdx.x * 16 + n;

    v2f a;                               // A matrix: all ones in every lane/VGPR
    a[0] = 1.0f;
    a[1] = 1.0f;

    const float inv_h = 1.0f / (float)H_;
    float p = 1.0f;

    #pragma unroll
    for (int l = 0; l < L_; ++l) {
        v8f c = {};                      // fresh accumulator per layer
        #pragma unroll
        for (int g = 0; g < 3; ++g) {    // 3 groups of 4 heads
            const int h0 = g * 4 + half;        // head for B vgpr0 in this lane half
            const int h1 = g * 4 + 2 + half;    // head for B vgpr1 in this lane half
            // att flat index: (((l*B + b)*H + h)*S + 0)*S + s
            const size_t plane = (size_t)(S_) * (size_t)(S_);
            const size_t base  = (size_t)((l * B_ + b) * H_) * plane;
            v2f bm;
            bm[0] = att[base + (size_t)h0 * plane + (size_t)s];
            bm[1] = att[base + (size_t)h1 * plane + (size_t)s];
            c = __builtin_amdgcn_wmma_f32_16x16x4_f32(
                    /*neg_a=*/false, a, /*neg_b=*/false, bm,
                    /*c_mod=*/(short)0, c, /*reuse_a=*/false, /*reuse_b=*/false);
        }
        // every D VGPR / every row holds colsum(n); take element 0
        p *= c[0] * inv_h;               // layer mean over 12 heads, f32 exact chain
    }

    if (half == 0) {                     // lanes 16-31 hold duplicate values
        ysoft[b * S_ + s] = -p;
    }
}

// Kernel 2: exact select_k matching the reference's double-argsort semantics.
// rank[s] = #{ s' : y[s'] < y[s]  ||  (y[s'] == y[s] && s' < s) }   (stable ties)
// y[0] is forced to (global_min(y_soft) - 1) before ranking, as in the reference.
__global__ __launch_bounds__(S_) void select_k_kernel(const float* __restrict__ ysoft,
                                                      float* __restrict__ yhard,
                                                      const float* __restrict__ rate_ptr) {
    __shared__ float row[S_];
    __shared__ float red[S_];
    const int t = threadIdx.x;

    // global min over the full [B,S] y_soft (pre-override values)
    float m = 3.4e38f;
    #pragma unroll
    for (int b = 0; b < B_; ++b) {
        m = fminf(m, ysoft[b * S_ + t]);
    }
    red[t] = m;
    __syncthreads();
    for (int st = S_ / 2; st > 0; st >>= 1) {
        if (t < st) red[t] = fminf(red[t], red[t + st]);
        __syncthreads();
    }
    const float gmin = red[0];

    // k = max(int(S * (1 - compression_rate)), 1), truncation toward zero
    const float cr = rate_ptr[0];
    int k = (int)((float)S_ * (1.0f - cr));
    if (k < 1) k = 1;

    for (int b = 0; b < B_; ++b) {
        float v = ysoft[b * S_ + t];
        if (t == 0) v = gmin - 1.0f;     // keep_first_unit: token 0 always kept
        row[t] = v;
        __syncthreads();

        int rank = 0;
        #pragma unroll 8
        for (int s = 0; s < S_; ++s) {
            const float w = row[s];
            rank += (w < v) || ((w == v) && (s < t));
        }
        yhard[b * S_ + t] = (rank < k) ? 1.0f : 0.0f;
        __syncthreads();
    }
}

extern "C" void kernel_launch(void* const* d_in, const int* in_sizes, int n_in,
                              void* d_out, int out_size, void* d_ws, size_t ws_size,
                              hipStream_t stream) {
    (void)in_sizes; (void)n_in; (void)d_ws; (void)ws_size; (void)out_size;

    const float* att  = (const float*)d_in[0];   // [L,B,H,S,S] fp32
    // d_in[1] (embedding_sequence) is dead: only its static shape S=512 is used
    const float* rate = (const float*)d_in[2];   // scalar compression_rate

    float* out   = (float*)d_out;
    float* yhard = out;                 // [B,S] as 0.0/1.0
    float* ysoft = out + B_ * S_;       // [B,S] fp32

    // 8 batches x 32 s-tiles of 16 -> 256 single-wave blocks
    ysoft_wmma_kernel<<<dim3(S_ / 16, B_), 32, 0, stream>>>(att, ysoft);
    // single 512-thread block does min-reduce + exact rank selection
    select_k_kernel<<<1, S_, 0, stream>>>(ysoft, yhard, rate);
}